// MDTA_17643725652098
// MI455X (gfx1250) — compile-verified
//
#include <hip/hip_runtime.h>

// MDTA for MI455X (gfx1250): memory-bound (~1.5 GB traffic @ 23.3 TB/s >> 35 GFLOP),
// fp32 throughout, V_WMMA_F32_16X16X4_F32 for all channel GEMMs, TDM for tile staging.

#define B_    2
#define C_    128
#define C3_   384
#define HDIM  48
#define N_    (HDIM * HDIM * HDIM)   // 110592
#define HEADS 8
#define DH    16
#define EPSN  1e-12f
#define CHUNK 512                     // Gram K-chunk per block (N_ % CHUNK == 0)
#define NT    64                      // n-tile for the two big GEMMs
#define LDSPITCH 72                   // NT + 8 pad -> halves 16 banks apart, conflict-free

typedef float v2f __attribute__((ext_vector_type(2)));
typedef float v8f __attribute__((ext_vector_type(8)));
typedef unsigned int u32x4 __attribute__((ext_vector_type(4)));
typedef int i32x8 __attribute__((ext_vector_type(8)));
typedef int i32x4 __attribute__((ext_vector_type(4)));

// WMMA f32 16x16x4 fragment layout (ISA 7.12.2, wave32):
//   A (16x4): lane L, vgpr v : A[M = L%16][K = 2*(L/16) + v]
//   B (4x16): lane L, vgpr r : B[K = r + 2*(L/16)][Ncol = L%16]
//   C/D     : lane L, vgpr r : C[M = r + 8*(L/16)][Ncol = L%16]

// -------- Tensor Data Mover: 2D tile (NT cols x rows) of f32, row stride = strideElems,
// -------- into LDS with +8-DWORD pad after every 64 DWORDs (row pitch 72 floats).
__device__ __forceinline__ void tdm_load_tile(const float* gptr, unsigned ldsOff,
                                              unsigned rows, unsigned tensorW,
                                              unsigned strideElems) {
  unsigned long long ga = (unsigned long long)(uintptr_t)gptr;
  u32x4 g0;
  g0[0] = 1u;                                        // count=1 (valid user D#)
  g0[1] = ldsOff;                                    // lds_addr (bytes)
  g0[2] = (unsigned)(ga & 0xFFFFFFFFu);              // global_addr[31:0]
  g0[3] = (unsigned)((ga >> 32) & 0x01FFFFFFu)       // global_addr[56:32]
        | (2u << 30);                                // type = 2 ("image")
  i32x8 g1;
  g1[0] = (int)((2u << 16)                           // data_size = 4 bytes
        | (1u << 20)                                 // pad_enable
        | (5u << 22)                                 // pad_interval: 64 DWORDs
        | (7u << 25));                               // pad_amount : 8 DWORDs
  g1[1] = (int)((tensorW & 0xFFFFu) << 16);          // tensor_dim0[15:0] (abar addr = 0)
  g1[2] = (int)(((tensorW >> 16) & 0xFFFFu)          // tensor_dim0[31:16]
        | ((rows & 0xFFFFu) << 16));                 // tensor_dim1[15:0]
  g1[3] = (int)((unsigned)NT << 16);                 // tensor_dim1[31:16]=0, tile_dim0=NT
  g1[4] = (int)(rows & 0xFFFFu);                     // tile_dim1 = rows, tile_dim2 = 0
  g1[5] = (int)strideElems;                          // tensor_dim0_stride[31:0]
  g1[6] = 0;                                         // stride hi, dim1_stride lo
  g1[7] = 0;
  i32x4 z4 = {0, 0, 0, 0};                           // groups 2/3 unused (2D)
  i32x8 z8 = {0, 0, 0, 0, 0, 0, 0, 0};
  __builtin_amdgcn_tensor_load_to_lds(g0, g1, z4, z4, z8, 0);
}

// ---------------- Kernel 1: qkv = qkv_w (384x128) @ x (128xN) ----------------
__global__ __launch_bounds__(256) void k_qkv_gemm(const float* __restrict__ x,
                                                  const float* __restrict__ w,
                                                  float* __restrict__ out) {
  const int b  = blockIdx.y;
  const int n0 = blockIdx.x * NT;
  __shared__ float xs[C_][LDSPITCH];
  const int t = threadIdx.x;
  const float* xb = x + (size_t)b * C_ * N_;
  if (t < 32) {                                      // wave 0 issues the DMA
    tdm_load_tile(xb + n0, (unsigned)(uintptr_t)&xs[0][0], C_, N_, N_);
    __builtin_amdgcn_s_wait_tensorcnt(0);
  }
  __syncthreads();
  const int wave = t >> 5, lane = t & 31;
  const int hf = lane >> 4, lm = lane & 15;
  float* ob = out + (size_t)b * C3_ * N_;
  for (int ot = wave; ot < C3_ / 16; ot += 8) {      // 24 o-tiles / 8 waves
    const int o0 = ot * 16;
    v8f acc[4] = {{}, {}, {}, {}};
    for (int k0 = 0; k0 < C_; k0 += 4) {
      v2f a = *(const v2f*)(w + (o0 + lm) * C_ + k0 + 2 * hf);  // reused 4x
      #pragma unroll
      for (int s = 0; s < 4; ++s) {                  // 4 independent WMMA chains
        v2f bf;
        bf.x = xs[k0 + 2 * hf][lm + 16 * s];
        bf.y = xs[k0 + 2 * hf + 1][lm + 16 * s];
        acc[s] = __builtin_amdgcn_wmma_f32_16x16x4_f32(false, a, false, bf,
                                                       (short)0, acc[s],
                                                       false, false);
      }
    }
    #pragma unroll
    for (int s = 0; s < 4; ++s)
      #pragma unroll
      for (int r = 0; r < 8; ++r) {
        int row = o0 + r + 8 * hf;
        ob[(size_t)row * N_ + n0 + 16 * s + lm] = acc[s][r];
      }
  }
}

// ---------------- Kernel 2: depthwise 3x3x3 conv, pad 1, 384 groups ----------
__global__ __launch_bounds__(256) void k_dwconv(const float* __restrict__ in,
                                                const float* __restrict__ wt,
                                                float* __restrict__ out) {
  const int bc = blockIdx.y;                   // b*384 + ch
  const int n  = blockIdx.x * 256 + threadIdx.x;
  const int ch = bc % C3_;
  const int z = n / (HDIM * HDIM);
  const int y = (n / HDIM) % HDIM;
  const int xx = n % HDIM;
  const float* ib = in + (size_t)bc * N_;
  const float* wc = wt + ch * 27;              // scalar-uniform per block
  float acc = 0.f;
  #pragma unroll
  for (int dz = -1; dz <= 1; ++dz) {
    int zz = z + dz;
    if (zz < 0 || zz >= HDIM) continue;
    #pragma unroll
    for (int dy = -1; dy <= 1; ++dy) {
      int yy = y + dy;
      if (yy < 0 || yy >= HDIM) continue;
      #pragma unroll
      for (int dx = -1; dx <= 1; ++dx) {
        int xc = xx + dx;
        if (xc < 0 || xc >= HDIM) continue;
        acc += wc[(dz + 1) * 9 + (dy + 1) * 3 + (dx + 1)] *
               ib[(zz * HDIM + yy) * HDIM + xc];
      }
    }
  }
  out[(size_t)bc * N_ + n] = acc;
}

// ------- Kernel 3: per-head Gram q*k^T over N (WMMA) + fused row sum-sq ------
__global__ __launch_bounds__(256) void k_gram(const float* __restrict__ qkv,
                                              float* __restrict__ gram,
                                              float* __restrict__ sqq,
                                              float* __restrict__ sqk) {
  const int bh = blockIdx.y;                   // b*8 + head
  const int b = bh >> 3, h = bh & 7;
  const int n0 = blockIdx.x * CHUNK;
  __shared__ float qs[16][CHUNK + 4];          // +4 pad: conflict-light col reads
  __shared__ float ks[16][CHUNK + 4];
  __shared__ float cred[8][256];
  __shared__ float rr[16][16];
  const int t = threadIdx.x;
  const float* qb = qkv + ((size_t)b * C3_ + h * DH) * N_;
  const float* kb = qkv + ((size_t)b * C3_ + C_ + h * DH) * N_;
  #pragma unroll
  for (int i = 0; i < 16 * CHUNK / 256; ++i) { // fully coalesced staging
    int idx = t + i * 256;
    int row = idx / CHUNK, col = idx % CHUNK;
    qs[row][col] = qb[(size_t)row * N_ + n0 + col];
    ks[row][col] = kb[(size_t)row * N_ + n0 + col];
  }
  __syncthreads();
  const int wave = t >> 5, lane = t & 31, hf = lane >> 4, lm = lane & 15;
  v8f acc = {};
  const int kbase = wave * (CHUNK / 8);        // split K across the 8 waves
  for (int kk = 0; kk < CHUNK / 8; kk += 4) {
    int nb = kbase + kk;
    v2f a, bf;
    a.x  = qs[lm][nb + 2 * hf];  a.y  = qs[lm][nb + 2 * hf + 1];
    bf.x = ks[lm][nb + 2 * hf];  bf.y = ks[lm][nb + 2 * hf + 1];  // B = k^T
    acc = __builtin_amdgcn_wmma_f32_16x16x4_f32(false, a, false, bf,
                                                (short)0, acc, false, false);
  }
  #pragma unroll
  for (int r = 0; r < 8; ++r) cred[wave][lane * 8 + r] = acc[r];
  __syncthreads();
  {                                            // cross-wave reduce, one atomic/elem
    float s = 0.f;
    #pragma unroll
    for (int wv = 0; wv < 8; ++wv) s += cred[wv][t];
    int lane2 = t >> 3, r2 = t & 7;
    int M = r2 + 8 * (lane2 >> 4), col = lane2 & 15;
    atomicAdd(&gram[bh * 256 + M * 16 + col], s);
  }
  // fused sum-of-squares for L2-normalize (separable row scaling of Gram)
  const int row = t >> 4, c0 = t & 15;
  float s1 = 0.f, s2 = 0.f;
  for (int col = c0; col < CHUNK; col += 16) {
    float v1 = qs[row][col]; s1 += v1 * v1;
    float v2 = ks[row][col]; s2 += v2 * v2;
  }
  __syncthreads();
  rr[row][c0] = s1;
  __syncthreads();
  if (t < 16) {
    float ss = 0.f;
    #pragma unroll
    for (int j = 0; j < 16; ++j) ss += rr[t][j];
    atomicAdd(&sqq[bh * 16 + t], ss);
  }
  __syncthreads();
  rr[row][c0] = s2;
  __syncthreads();
  if (t < 16) {
    float ss = 0.f;
    #pragma unroll
    for (int j = 0; j < 16; ++j) ss += rr[t][j];
    atomicAdd(&sqk[bh * 16 + t], ss);
  }
}

// -------- Kernel 4a: normalize + temperature + 16-wide softmax ---------------
__global__ __launch_bounds__(256) void k_softmax(const float* __restrict__ gram,
                                                 const float* __restrict__ sqq,
                                                 const float* __restrict__ sqk,
                                                 const float* __restrict__ temp,
                                                 float* __restrict__ P) {
  const int bh = blockIdx.x, h = bh & 7;
  const int t = threadIdx.x, i = t >> 4, j = t & 15;
  float rq = 1.f / fmaxf(sqrtf(sqq[bh * 16 + i]), EPSN);
  float rk = 1.f / fmaxf(sqrtf(sqk[bh * 16 + j]), EPSN);
  float s = gram[bh * 256 + i * 16 + j] * rq * rk * temp[h];
  float m = s;
  #pragma unroll
  for (int off = 8; off; off >>= 1) m = fmaxf(m, __shfl_xor(m, off, 16));
  float e = __expf(s - m);
  float sum = e;
  #pragma unroll
  for (int off = 8; off; off >>= 1) sum += __shfl_xor(sum, off, 16);
  P[bh * 256 + i * 16 + j] = e / sum;
}

// -------- Kernel 4b: W_eff[b] = proj_w @ blockdiag(P[b])  (fuses proj) -------
__global__ __launch_bounds__(256) void k_weff(const float* __restrict__ projw,
                                              const float* __restrict__ P,
                                              float* __restrict__ weff) {
  int idx = blockIdx.x * 256 + threadIdx.x;    // [b][o][c'] flat, 32768 total
  int b = idx >> 14;
  int o = (idx >> 7) & 127;
  int cp = idx & 127;
  int h = cp >> 4, j = cp & 15;
  const float* pw = projw + o * C_ + h * 16;
  const float* pp = P + ((b * 8 + h) * 16) * 16 + j;
  float s = 0.f;
  #pragma unroll
  for (int i = 0; i < 16; ++i) s += pw[i] * pp[i * 16];
  weff[idx] = s;
}

// -------- Kernel 5: y = W_eff[b] (128x128) @ v (128xN) -----------------------
__global__ __launch_bounds__(256) void k_out_gemm(const float* __restrict__ qkv,
                                                  const float* __restrict__ weff,
                                                  float* __restrict__ out) {
  const int b  = blockIdx.y;
  const int n0 = blockIdx.x * NT;
  __shared__ float vs[C_][LDSPITCH];
  const int t = threadIdx.x;
  const float* vb = qkv + ((size_t)b * C3_ + 2 * C_) * N_;  // v = rows 256..383
  if (t < 32) {
    tdm_load_tile(vb + n0, (unsigned)(uintptr_t)&vs[0][0], C_, N_, N_);
    __builtin_amdgcn_s_wait_tensorcnt(0);
  }
  __syncthreads();
  const int wave = t >> 5, lane = t & 31, hf = lane >> 4, lm = lane & 15;
  const int o0 = wave * 16;                    // 8 o-tiles == 8 waves
  const float* wb = weff + b * C_ * C_;
  v8f acc[4] = {{}, {}, {}, {}};
  for (int k0 = 0; k0 < C_; k0 += 4) {
    v2f a = *(const v2f*)(wb + (o0 + lm) * C_ + k0 + 2 * hf);
    #pragma unroll
    for (int s = 0; s < 4; ++s) {
      v2f bf;
      bf.x = vs[k0 + 2 * hf][lm + 16 * s];
      bf.y = vs[k0 + 2 * hf + 1][lm + 16 * s];
      acc[s] = __builtin_amdgcn_wmma_f32_16x16x4_f32(false, a, false, bf,
                                                     (short)0, acc[s],
                                                     false, false);
    }
  }
  float* ob = out + (size_t)b * C_ * N_;
  #pragma unroll
  for (int s = 0; s < 4; ++s)
    #pragma unroll
    for (int r = 0; r < 8; ++r) {
      int row = o0 + r + 8 * hf;
      ob[(size_t)row * N_ + n0 + 16 * s + lm] = acc[s][r];
    }
}

extern "C" void kernel_launch(void* const* d_in, const int* in_sizes, int n_in,
                              void* d_out, int out_size, void* d_ws, size_t ws_size,
                              hipStream_t stream) {
  (void)in_sizes; (void)n_in; (void)out_size; (void)ws_size;
  const float* x      = (const float*)d_in[0];
  const float* qkv_w  = (const float*)d_in[1];
  const float* dw_w   = (const float*)d_in[2];
  const float* proj_w = (const float*)d_in[3];
  const float* temp   = (const float*)d_in[4];
  float* out = (float*)d_out;

  float* ws = (float*)d_ws;
  const size_t qkvN = (size_t)B_ * C3_ * N_;   // 84,934,656 floats
  float* qkv0 = ws;
  float* qkv1 = qkv0 + qkvN;
  float* gram = qkv1 + qkvN;                   // 2*8*256
  float* sqq  = gram + (size_t)B_ * HEADS * 256;
  float* sqk  = sqq + B_ * HEADS * 16;
  float* P    = sqk + B_ * HEADS * 16;
  float* weff = P + (size_t)B_ * HEADS * 256;

  // zero the atomic accumulators (gram + sqq + sqk are contiguous)
  (void)hipMemsetAsync(gram, 0,
                       (size_t)(B_ * HEADS * 256 + 2 * B_ * HEADS * 16) *
                           sizeof(float),
                       stream);

  dim3 blk(256);
  k_qkv_gemm<<<dim3(N_ / NT, B_), blk, 0, stream>>>(x, qkv_w, qkv0);
  k_dwconv <<<dim3(N_ / 256, B_ * C3_), blk, 0, stream>>>(qkv0, dw_w, qkv1);
  k_gram   <<<dim3(N_ / CHUNK, B_ * HEADS), blk, 0, stream>>>(qkv1, gram, sqq, sqk);
  k_softmax<<<dim3(B_ * HEADS), blk, 0, stream>>>(gram, sqq, sqk, temp, P);
  k_weff   <<<dim3(B_ * C_ * C_ / 256), blk, 0, stream>>>(proj_w, P, weff);
  k_out_gemm<<<dim3(N_ / NT, B_), blk, 0, stream>>>(qkv1, weff, out);
}